// RGCNScratch_46866683134591
// MI455X (gfx1250) — compile-verified
//
#include <hip/hip_runtime.h>

// Problem constants (fixed by the reference harness)
#define N_NODES  50000
#define N_EDGES  800000
#define N_REL    8
#define HID      256
#define N_BASES  30
#define N_LAYERS 2

typedef float v2f __attribute__((ext_vector_type(2)));
typedef float v4f __attribute__((ext_vector_type(4)));
typedef float v8f __attribute__((ext_vector_type(8)));
typedef int   v4i __attribute__((ext_vector_type(4)));

#define AS1 __attribute__((address_space(1)))
#define AS3 __attribute__((address_space(3)))

#if defined(__gfx1250__) && __has_builtin(__builtin_amdgcn_global_load_async_to_lds_b128)
#define USE_ASYNC_LDS 1
#else
#define USE_ASYNC_LDS 0
#endif

// ---------------------------------------------------------------- utilities
__global__ void k_zero(float* __restrict__ p, int n) {
  int i = blockIdx.x * blockDim.x + threadIdx.x;
  int stride = gridDim.x * blockDim.x;
  for (; i < n; i += stride) p[i] = 0.0f;
}

__global__ void k_deg_count(const int* __restrict__ dst, float* __restrict__ deg) {
  int i = blockIdx.x * blockDim.x + threadIdx.x;
  int stride = gridDim.x * blockDim.x;
  for (; i < N_EDGES; i += stride) {
    unsafeAtomicAdd(&deg[dst[i]], 1.0f);   // -> global_atomic_add_f32
  }
}

__global__ void k_deg_inv(float* __restrict__ deg) {
  int i = blockIdx.x * blockDim.x + threadIdx.x;
  if (i < N_NODES) deg[i] = 1.0f / fmaxf(deg[i], 1.0f);
}

// W[r,i,o] = sum_b coeffs[r,b] * bases[b,i,o]   (per layer)
__global__ void k_wcalc(const float* __restrict__ bases,
                        const float* __restrict__ coeffs,
                        float* __restrict__ W) {
  int idx = blockIdx.x * blockDim.x + threadIdx.x;   // exactly R*H*H threads
  int r  = idx >> 16;           // H*H = 65536
  int io = idx & 65535;
  float s = 0.0f;
#pragma unroll
  for (int b = 0; b < N_BASES; ++b)
    s += coeffs[r * N_BASES + b] * bases[(size_t)b * (HID * HID) + io];
  W[idx] = s;
}

// Aggregate-first message passing for one relation:
//   agg[dst,:] += deg_inv[dst] * x[src,:]   for edges with edge_type == rel
// One wave per edge; 32 lanes move 256 floats (8 each).
__global__ void k_scatter(const int* __restrict__ src, const int* __restrict__ dst,
                          const int* __restrict__ etype, int rel,
                          const float* __restrict__ x,
                          const float* __restrict__ deg_inv,
                          float* __restrict__ agg) {
  int lane  = threadIdx.x & 31;
  int wave  = (blockIdx.x * blockDim.x + threadIdx.x) >> 5;
  int nwave = (gridDim.x * blockDim.x) >> 5;
  for (int e = wave; e < N_EDGES; e += nwave) {
    if (etype[e] != rel) continue;
    int s = src[e], d = dst[e];
    float w = deg_inv[d];
    const float* xs = x + (size_t)s * HID;
    float* ad = agg + (size_t)d * HID;
#pragma unroll
    for (int j = 0; j < 8; ++j) {
      int c = lane + (j << 5);                         // coalesced across the wave
      unsafeAtomicAdd(&ad[c], xs[c] * w);              // L2-resident atomics (agg = 51MB < 192MB L2)
    }
  }
}

// C[M x 256] (+)= A[M x 256] @ B[256 x 256] using V_WMMA_F32_16X16X4_F32.
// Block: 256 threads = 8 waves. Each wave: 16 rows x 64 cols (4 accumulator tiles).
// Block tile: 128 rows x 64 cols; grid = (ceil(M/128), 256/64).
// The B panel (256 K x 64 cols) is shared by all 8 waves -> staged once in LDS
// via the CDNA5 async load-to-LDS path (ASYNCcnt) when available.
#define BSTR 72   // LDS row stride (floats): 2*72 % 64 == 16 -> lane halves hit
                  // disjoint bank ranges [b,b+15] vs [b+16,b+31] (conflict-free)

__global__ __launch_bounds__(256)
void k_gemm(const float* __restrict__ A, const float* __restrict__ B,
            float* __restrict__ C, int M, int accum, int relu) {
  __shared__ float Bs[256 * BSTR];   // 73.7 KB of the WGP's 320 KB LDS

  const int tid  = threadIdx.x;
  const int wave = tid >> 5;
  const int lane = tid & 31;
  const int hi   = lane >> 4;      // 0 or 1 (lane half)
  const int lo   = lane & 15;

  const int row0 = blockIdx.x * 128 + wave * 16;   // wave's 16-row strip
  const int col0 = blockIdx.y * 64;                // block's 64-col strip

  // ---- stage B[:, col0:col0+64) into LDS: 4096 x float4, 16 per thread ----
#pragma unroll
  for (int i = 0; i < 16; ++i) {
    const int q    = tid + i * 256;        // 0..4095
    const int row  = q >> 4;               // 0..255 (K index)
    const int quad = (q & 15) << 2;        // 0,4,...,60 (col within panel)
    const float* gp = B + (size_t)row * HID + col0 + quad;
    float*       lp = &Bs[row * BSTR + quad];
#if USE_ASYNC_LDS
    __builtin_amdgcn_global_load_async_to_lds_b128(
        (AS1 v4i*)gp, (AS3 v4i*)lp, 0, 0);
#else
    *(v4f*)lp = *(const v4f*)gp;
#endif
  }
#if USE_ASYNC_LDS
#if __has_builtin(__builtin_amdgcn_s_wait_asynccnt)
  __builtin_amdgcn_s_wait_asynccnt(0);
#else
  asm volatile("s_wait_asynccnt 0x0" ::: "memory");
#endif
#endif
  __syncthreads();

  // A-fragment row for this lane (clamped so OOB rows read row M-1; those
  // rows' results are never stored, so correctness is unaffected).
  int ra = row0 + lo;
  if (ra > M - 1) ra = M - 1;
  const float* Arow = A + (size_t)ra * HID;

  v8f acc[4];
#pragma unroll
  for (int t = 0; t < 4; ++t) acc[t] = (v8f){};

#pragma unroll 4
  for (int k0 = 0; k0 < HID; k0 += 4) {
    const int ka = k0 + 2 * hi;                    // this lane-half's K pair
    // A 16x4 fragment: lanes 0-15 -> K {k0,k0+1}, lanes 16-31 -> K {k0+2,k0+3}
    v2f a = *(const v2f*)(Arow + ka);              // 8B-aligned (ka even)

    const float* bL = &Bs[ka * BSTR + lo];         // LDS row ka, this lane's col
    v2f b;

    b.x = bL[0];       b.y = bL[BSTR + 0];
    acc[0] = __builtin_amdgcn_wmma_f32_16x16x4_f32(false, a, false, b, (short)0, acc[0], false, false);
    b.x = bL[16];      b.y = bL[BSTR + 16];
    acc[1] = __builtin_amdgcn_wmma_f32_16x16x4_f32(false, a, false, b, (short)0, acc[1], false, false);
    b.x = bL[32];      b.y = bL[BSTR + 32];
    acc[2] = __builtin_amdgcn_wmma_f32_16x16x4_f32(false, a, false, b, (short)0, acc[2], false, false);
    b.x = bL[48];      b.y = bL[BSTR + 48];
    acc[3] = __builtin_amdgcn_wmma_f32_16x16x4_f32(false, a, false, b, (short)0, acc[3], false, false);
  }

  // Epilogue: C/D layout — VGPR v holds row (v + 8*hi), col = lane&15.
#pragma unroll
  for (int t = 0; t < 4; ++t) {
    const int col = col0 + t * 16 + lo;
#pragma unroll
    for (int v = 0; v < 8; ++v) {
      const int row = row0 + v + 8 * hi;
      if (row < M) {
        const size_t idx = (size_t)row * HID + col;
        float val = acc[t][v];
        if (accum) val += C[idx];
        if (relu)  val = fmaxf(val, 0.0f);
        C[idx] = val;
      }
    }
  }
}

// ---------------------------------------------------------------- launcher
extern "C" void kernel_launch(void* const* d_in, const int* in_sizes, int n_in,
                              void* d_out, int out_size, void* d_ws, size_t ws_size,
                              hipStream_t stream) {
  const int*   edge_index = (const int*)d_in[0];    // [2, E]
  const int*   src        = edge_index;
  const int*   dst        = edge_index + N_EDGES;
  const int*   etype      = (const int*)d_in[1];    // [E]
  const float* emb        = (const float*)d_in[3];  // [N, H]
  const float* bases      = (const float*)d_in[4];  // [L, B, H, H]
  const float* coeffs     = (const float*)d_in[5];  // [L, R, B]
  const float* self_loops = (const float*)d_in[6];  // [L, H, H]
  float*       out        = (float*)d_out;          // [N, H]

  // Workspace layout (floats): W | deg_inv | agg | xbuf   (~105 MB total)
  float* W       = (float*)d_ws;                          // R*H*H   = 524288
  float* deg_inv = W + (size_t)N_REL * HID * HID;         // N (padded)
  float* agg     = deg_inv + 50176;                       // N*H = 12.8M
  float* xbuf    = agg + (size_t)N_NODES * HID;           // N*H = 12.8M

  // In-degree normalization (recomputed every call: deterministic)
  k_zero<<<256, 256, 0, stream>>>(deg_inv, N_NODES);
  k_deg_count<<<512, 256, 0, stream>>>(dst, deg_inv);
  k_deg_inv<<<(N_NODES + 255) / 256, 256, 0, stream>>>(deg_inv);

  dim3 ggrid((N_NODES + 127) / 128, HID / 64);

  for (int l = 0; l < N_LAYERS; ++l) {
    const float* xin  = (l == 0) ? emb  : xbuf;
    float*       xout = (l == 0) ? xbuf : out;

    // W[r] = sum_b coeff[r,b] * bases[b]
    k_wcalc<<<(N_REL * HID * HID) / 256, 256, 0, stream>>>(
        bases + (size_t)l * N_BASES * HID * HID, coeffs + (size_t)l * N_REL * N_BASES, W);

    // xout = xin @ self_loops[l]            (initializes xout)
    k_gemm<<<ggrid, 256, 0, stream>>>(xin, self_loops + (size_t)l * HID * HID,
                                      xout, N_NODES, /*accum=*/0, /*relu=*/0);

    for (int r = 0; r < N_REL; ++r) {
      // agg = segment_sum over edges of relation r of deg_inv[dst]*xin[src]
      k_zero<<<2048, 256, 0, stream>>>(agg, N_NODES * HID);
      k_scatter<<<1024, 256, 0, stream>>>(src, dst, etype, r, xin, deg_inv, agg);
      // xout += agg @ W[r]  ; relu fused into the final accumulation of layer 0
      const int relu = (l == 0 && r == N_REL - 1) ? 1 : 0;
      k_gemm<<<ggrid, 256, 0, stream>>>(agg, W + (size_t)r * HID * HID,
                                        xout, N_NODES, /*accum=*/1, relu);
    }
  }
}